// PPONet_39883066310844
// MI455X (gfx1250) — compile-verified
//
#include <hip/hip_runtime.h>

typedef __attribute__((ext_vector_type(2))) float v2f;
typedef __attribute__((ext_vector_type(8))) float v8f;

#define NB 16
#define NN 256
#define ND 8
#define NE 64
#define NT 5
#define MUS 68   // padded row stride of mu (floats): conflict-free LDS A-frags

// One workgroup per batch. 512 threads = 16 waves (wave32); wave w owns rows
// [16w, 16w+16) of the 256xEMB state. Whole pipeline runs in one kernel.
// Every 256x64 @ 64x64 GEMM (w1b, w3, w2, w7) runs on V_WMMA_F32_16X16X4_F32
// against a time-multiplexed K-paired 16KB LDS weight buffer `wp`;
// the big per-iteration s2 = W @ (mu@w2) GEMM streams A-frags from global W.
__launch_bounds__(512, 1)
__global__ void s2v_fused_kernel(const float* __restrict__ x,
                                 const float* __restrict__ Wg,
                                 const unsigned char* __restrict__ reach,
                                 const float* __restrict__ w1a, const float* __restrict__ b1a,
                                 const float* __restrict__ w1b, const float* __restrict__ b1b,
                                 const float* __restrict__ w2,  const float* __restrict__ b2,
                                 const float* __restrict__ w3,  const float* __restrict__ b3,
                                 const float* __restrict__ w4,  const float* __restrict__ b4,
                                 const float* __restrict__ w5,  const float* __restrict__ b5,
                                 const float* __restrict__ w6,  const float* __restrict__ b6,
                                 const float* __restrict__ w7,  const float* __restrict__ b7,
                                 float* __restrict__ out)
{
    __shared__ float mu[NN * MUS];        // ~68 KB  state / staging (padded rows)
    __shared__ float pre[NN * NE];        //  64 KB  s1 + s3
    __shared__ float muw[(NN / 2) * 128]; //  64 KB  mu@w2 K-paired; later reduce scratch
    __shared__ float wp[(NE / 2) * 128];  //  16 KB  current 64x64 weight, K-paired
    __shared__ float bs[NE];              //  current bias
    __shared__ float gs[NE];
    __shared__ float gst[NE];
    __shared__ float gsc;
    __shared__ float red[512];

    const int b   = blockIdx.x;
    const int tid = threadIdx.x;
    const float* __restrict__ Wb = Wg + (size_t)b * NN * NN;
    const float* __restrict__ xb = x  + (size_t)b * NN * ND;

    const int wv   = tid >> 5;
    const int lane = tid & 31;
    const int half = lane >> 4;   // 0: lanes 0-15, 1: lanes 16-31
    const int r    = lane & 15;
    const int m0   = wv * 16;

    // wave-level GEMM: acc[4] += mu[rows m0..m0+15, 0..63] @ wp (K=64)
    auto gemm_mu_wp = [&](v8f* acc) {
#pragma unroll 4
        for (int k0 = 0; k0 < NE; k0 += 4) {
            v2f a = *(const v2f*)(mu + (m0 + r) * MUS + k0 + 2 * half);
            const int kp = ((k0 + 2 * half) >> 1) * 128;
#pragma unroll
            for (int t4 = 0; t4 < 4; ++t4) {
                v2f bf = *(const v2f*)(wp + kp + (t4 * 16 + r) * 2);
                acc[t4] = __builtin_amdgcn_wmma_f32_16x16x4_f32(
                    false, a, false, bf, (short)0, acc[t4], false, false);
            }
        }
    };
    // pack a row-major 64x64 global matrix + bias into wp/bs (K-paired)
    auto pack_wp = [&](const float* __restrict__ wsrc, const float* __restrict__ bsrc) {
        for (int idx = tid; idx < NE * NE; idx += 512) {
            const int k = idx >> 6, e = idx & 63;
            wp[(k >> 1) * 128 + e * 2 + (k & 1)] = wsrc[k * NE + e];
        }
        if (tid < NE) bs[tid] = bsrc[tid];
    };

    // ---- Phase A: pre = s1 + s3 -----------------------------------------
    // region0: t1 = relu(x @ w1a + b1a) -> mu (staging); pack w1b
    pack_wp(w1b, b1b);
    for (int idx = tid; idx < NN * NE; idx += 512) {
        const int n = idx >> 6, e = idx & 63;
        float t = b1a[e];
#pragma unroll
        for (int k = 0; k < ND; ++k) t += xb[n * ND + k] * w1a[k * NE + e];
        mu[n * MUS + e] = fmaxf(t, 0.f);
    }
    __syncthreads();
    // region1: pre = t1 @ w1b + b1b   (WMMA)
    {
        v8f acc[4] = {};
        gemm_mu_wp(acc);
#pragma unroll
        for (int t4 = 0; t4 < 4; ++t4)
#pragma unroll
            for (int r8 = 0; r8 < 8; ++r8) {
                const int row = m0 + r8 + 8 * half;
                const int col = t4 * 16 + r;
                pre[row * NE + col] = acc[t4][r8] + bs[col];
            }
    }
    __syncthreads();
    // region2: s3_2[n][e] = sum_i relu(W[i][n]*w4[e]+b4[e]) -> mu; pack w3
    pack_wp(w3, b3);
    for (int idx = tid; idx < NN * NE; idx += 512) {
        const int n = idx >> 6, e = idx & 63;
        const float w4e = w4[e], b4e = b4[e];
        float s = 0.f;
        for (int i = 0; i < NN; ++i) s += fmaxf(Wb[i * NN + n] * w4e + b4e, 0.f);
        mu[n * MUS + e] = s;
    }
    __syncthreads();
    // region3: pre += s3_2 @ w3 + b3   (WMMA)
    {
        v8f acc[4] = {};
        gemm_mu_wp(acc);
#pragma unroll
        for (int t4 = 0; t4 < 4; ++t4)
#pragma unroll
            for (int r8 = 0; r8 < 8; ++r8) {
                const int row = m0 + r8 + 8 * half;
                const int col = t4 * 16 + r;
                pre[row * NE + col] += acc[t4][r8] + bs[col];
            }
    }
    __syncthreads();
    // region4: mu = 0; pack w2
    pack_wp(w2, b2);
    for (int idx = tid; idx < NN * MUS; idx += 512) mu[idx] = 0.f;
    __syncthreads();

    // ---- Phase B: T message-passing iterations --------------------------
    for (int it = 0; it < NT; ++it) {
        // muw = mu @ w2   (K = 64, WMMA), stored K-paired
        v8f macc[4] = {};
        gemm_mu_wp(macc);
#pragma unroll
        for (int t4 = 0; t4 < 4; ++t4)
#pragma unroll
            for (int r8 = 0; r8 < 8; ++r8) {
                const int row = m0 + r8 + 8 * half;
                const int col = t4 * 16 + r;
                muw[(row >> 1) * 128 + col * 2 + (row & 1)] = macc[t4][r8];
            }
        __syncthreads();   // muw complete everywhere

        // s2 = W[rows m0..m0+15] @ muw   (K = 256, WMMA)
        v8f sacc[4] = {};
#pragma unroll 4
        for (int k0 = 0; k0 < NN; k0 += 4) {
            v2f a = *(const v2f*)(Wb + (m0 + r) * NN + k0 + 2 * half);
            const int kp = ((k0 + 2 * half) >> 1) * 128;
#pragma unroll
            for (int t4 = 0; t4 < 4; ++t4) {
                v2f bf = *(const v2f*)(muw + kp + (t4 * 16 + r) * 2);
                sacc[t4] = __builtin_amdgcn_wmma_f32_16x16x4_f32(
                    false, a, false, bf, (short)0, sacc[t4], false, false);
            }
        }
        // mu = relu(pre + s2 + b2)  (own rows; no cross-wave hazard on mu)
#pragma unroll
        for (int t4 = 0; t4 < 4; ++t4)
#pragma unroll
            for (int r8 = 0; r8 < 8; ++r8) {
                const int row = m0 + r8 + 8 * half;
                const int col = t4 * 16 + r;
                mu[row * MUS + col] =
                    fmaxf(pre[row * NE + col] + sacc[t4][r8] + bs[col], 0.f);
            }
        __syncthreads();   // muw free for next iteration / mu final
    }

    // ---- Phase C: readout ------------------------------------------------
    // pack w7 for the local-action GEMM; gsum partials in parallel
    pack_wp(w7, b7);
    {
        const int e = tid & 63, seg = tid >> 6;
        float s = 0.f;
        for (int n = seg * 32; n < seg * 32 + 32; ++n) s += mu[n * MUS + e];
        red[tid] = s;
    }
    __syncthreads();
    if (tid < NE) {
        float s = 0.f;
#pragma unroll
        for (int seg = 0; seg < 8; ++seg) s += red[seg * NE + tid];
        gs[tid] = s;
    }
    __syncthreads();
    if (tid < NE) {
        float g = b6[tid];
        for (int j = 0; j < NE; ++j) g += gs[j] * w6[j * NE + tid];
        gst[tid] = fmaxf(g, 0.f);   // relu of global-state half of rep
    }
    __syncthreads();
    if (tid == 0) {
        float s = b5[0];
        for (int e = 0; e < NE; ++e) s += gst[e] * w5[e];
        gsc = s;                    // global contribution, shared by all n
    }
    __syncthreads();

    // local half via WMMA: la = mu @ w7; logit[n] = gsc + sum_e relu(la+b7)*w5[64+e]
    {
        v8f la[4] = {};
        gemm_mu_wp(la);
        float* sc = muw + wv * 256;   // per-wave reduce scratch (muw is free)
#pragma unroll
        for (int r8 = 0; r8 < 8; ++r8) {
            float p = 0.f;
#pragma unroll
            for (int t4 = 0; t4 < 4; ++t4) {
                const int col = t4 * 16 + r;
                p += fmaxf(la[t4][r8] + bs[col], 0.f) * w5[NE + col];
            }
            sc[(r8 + 8 * half) * 16 + r] = p;
        }
        __builtin_amdgcn_wave_barrier();   // wave-local LDS store->load ordering
        if (lane < 16) {
            float s = gsc;
#pragma unroll
            for (int q = 0; q < 16; ++q) s += sc[lane * 16 + q];
            const int n = m0 + lane;
            out[b * NN + n] = reach[b * NN + n] ? s : -__builtin_inff();
        }
    }
}

extern "C" void kernel_launch(void* const* d_in, const int* in_sizes, int n_in,
                              void* d_out, int out_size, void* d_ws, size_t ws_size,
                              hipStream_t stream) {
    (void)in_sizes; (void)n_in; (void)d_ws; (void)ws_size; (void)out_size;
    const float* x    = (const float*)d_in[0];
    const float* W    = (const float*)d_in[1];
    const unsigned char* reach = (const unsigned char*)d_in[2];
    const float* w1a  = (const float*)d_in[3];
    const float* b1a  = (const float*)d_in[4];
    const float* w1b  = (const float*)d_in[5];
    const float* b1b  = (const float*)d_in[6];
    const float* w2   = (const float*)d_in[7];
    const float* b2   = (const float*)d_in[8];
    const float* w3   = (const float*)d_in[9];
    const float* b3   = (const float*)d_in[10];
    const float* w4   = (const float*)d_in[11];
    const float* b4   = (const float*)d_in[12];
    const float* w5   = (const float*)d_in[13];
    const float* b5   = (const float*)d_in[14];
    const float* w6   = (const float*)d_in[15];
    const float* b6   = (const float*)d_in[16];
    const float* w7   = (const float*)d_in[17];
    const float* b7   = (const float*)d_in[18];
    float* out = (float*)d_out;

    s2v_fused_kernel<<<dim3(NB), dim3(512), 0, stream>>>(
        x, W, reach, w1a, b1a, w1b, b1b, w2, b2, w3, b3, w4, b4,
        w5, b5, w6, b6, w7, b7, out);
}